// Downsampler_74491912782199
// MI455X (gfx1250) — compile-verified
//
#include <hip/hip_runtime.h>
#include <math.h>

#define K2    9
#define B_    8
#define H_    512
#define W_    512
#define h_    256
#define w_    256
#define NSEG  45            // 9 kernel segs + 18 offx segs + 18 offy segs
#define SEGF  256           // floats per segment (one row)

// ---------------------------------------------------------------------------
// gfx1250 async global->LDS path (guarded; sync fallback keeps compile clean)
// ---------------------------------------------------------------------------
#if defined(__gfx1250__) && __has_builtin(__builtin_amdgcn_global_load_async_to_lds_b32)
#define HAVE_ASYNC_LDS 1
#else
#define HAVE_ASYNC_LDS 0
#endif

#if HAVE_ASYNC_LDS
typedef int v4i __attribute__((vector_size(16)));
typedef __attribute__((address_space(1))) int* g_i32p;
typedef __attribute__((address_space(3))) int* l_i32p;
typedef __attribute__((address_space(1))) v4i* g_v4ip;
typedef __attribute__((address_space(3))) v4i* l_v4ip;
#endif

__device__ __forceinline__ void cp_g2lds_b128(const float* g, float* l) {
#if HAVE_ASYNC_LDS
#if __has_builtin(__builtin_amdgcn_global_load_async_to_lds_b128)
    __builtin_amdgcn_global_load_async_to_lds_b128((g_v4ip)g, (l_v4ip)l, 0, 0);
#else
    __builtin_amdgcn_global_load_async_to_lds_b32((g_i32p)(g + 0), (l_i32p)(l + 0), 0, 0);
    __builtin_amdgcn_global_load_async_to_lds_b32((g_i32p)(g + 1), (l_i32p)(l + 1), 0, 0);
    __builtin_amdgcn_global_load_async_to_lds_b32((g_i32p)(g + 2), (l_i32p)(l + 2), 0, 0);
    __builtin_amdgcn_global_load_async_to_lds_b32((g_i32p)(g + 3), (l_i32p)(l + 3), 0, 0);
#endif
#else
    l[0] = g[0]; l[1] = g[1]; l[2] = g[2]; l[3] = g[3];
#endif
}

__device__ __forceinline__ void wait_async_lds() {
#if HAVE_ASYNC_LDS
#if __has_builtin(__builtin_amdgcn_s_wait_asynccnt)
    __builtin_amdgcn_s_wait_asynccnt(0);
#else
    asm volatile("s_wait_asynccnt 0" ::: "memory");
#endif
#endif
}

// ---------------------------------------------------------------------------
// Pass 1: [B,3,512,512] CHW -> [B,512,512] float4 channel-last (pad .w = 0)
// so each bilinear corner becomes a single b128 load in pass 2.
// ---------------------------------------------------------------------------
__global__ __launch_bounds__(256) void chw_to_hwc4(const float* __restrict__ img,
                                                   float4* __restrict__ out) {
    int idx = blockIdx.x * blockDim.x + threadIdx.x;     // over B*H*W
    if (idx >= B_ * H_ * W_) return;
    int b  = idx >> 18;                                  // / (512*512)
    int hw = idx & (H_ * W_ - 1);
    const float* p = img + (size_t)b * 3 * H_ * W_ + hw;
    float4 v;
    v.x = p[0];
    v.y = p[(size_t)H_ * W_];
    v.z = p[(size_t)2 * H_ * W_];
    v.w = 0.0f;
    out[idx] = v;
}

// ---------------------------------------------------------------------------
// Pass 2: one block per (b, output-row i); 256 lanes = output columns j.
// Faithful scrambled-bilinear semantics of the reference:
//   weights for sample n=(p=i*256+j, k) come from samples 2n, 2n+1 of the
//   (1-frac | frac) half selected by i<128; corner coords are exact integers
//   x0 = j + k/3 + 2 (row), y0 = j + k%3 + 2 (col); frac == offset value.
//
// LDS segment map (each segment = one 256-float row):
//   seg 0..8   : kernels row i, tap k=seg
//   seg 9..26  : offx rows, tap k=(seg-9)/2, row ibase+((seg-9)&1)
//   seg 27..44 : offy rows, tap k=(seg-27)/2, row ibase+((seg-27)&1)
// ---------------------------------------------------------------------------
template <bool USE_CL>
__global__ __launch_bounds__(256) void deform_gather(
    const float4* __restrict__ imcl,   // [B,512,512] float4 (USE_CL)
    const float*  __restrict__ imchw,  // [B,3,512,512]      (!USE_CL)
    const float*  __restrict__ kern,   // [B,9,256,256]
    const float*  __restrict__ offx,   // [B,9,256,256]
    const float*  __restrict__ offy,   // [B,9,256,256]
    float*        __restrict__ out)    // [B,256,256,3]
{
    __shared__ float sBuf[NSEG * SEGF];                 // 45 KB

    const int  tid   = threadIdx.x;          // j
    const int  i     = blockIdx.x & (h_ - 1);
    const int  b     = blockIdx.x >> 8;
    const bool lower = (i < h_ / 2);
    const int  ibase = lower ? (2 * i) : (2 * (i - h_ / 2));

    // ---- phase 1: issue async staging (b128, flattened over segments) ----
    const size_t bOff = (size_t)b * K2 * h_ * w_;
    for (int idx = tid; idx < NSEG * (SEGF / 4); idx += 256) {
        const int seg  = idx >> 6;           // /64 quads per segment
        const int quad = idx & 63;
        const float* src;
        if (seg < 9) {
            src = kern + bOff + ((size_t)seg * h_ + i) * w_;
        } else if (seg < 27) {
            const int s2 = seg - 9;
            src = offx + bOff + ((size_t)(s2 >> 1) * h_ + ibase + (s2 & 1)) * w_;
        } else {
            const int s2 = seg - 27;
            src = offy + bOff + ((size_t)(s2 >> 1) * h_ + ibase + (s2 & 1)) * w_;
        }
        cp_g2lds_b128(src + quad * 4, &sBuf[idx * 4]);
    }

    // ---- phase 2: issue all 36 corner gathers (independent of LDS) ----
    float4 Ia[K2], Ib[K2], Ic[K2], Id[K2];   // a=(x0,y0) b=(x0,y1) c=(x1,y0) d=(x1,y1)
    #pragma unroll
    for (int k = 0; k < K2; ++k) {
        const int x0 = tid + (k / 3) + 2;    // row into H  (in [2,260])
        const int y0 = tid + (k % 3) + 2;    // col into W
        if (USE_CL) {
            const float4* p = imcl + ((size_t)b * H_ + x0) * W_ + y0;
            Ia[k] = p[0]; Ib[k] = p[1]; Ic[k] = p[W_]; Id[k] = p[W_ + 1];
        } else {
            const size_t cs = (size_t)H_ * W_;
            const float* p = imchw + (size_t)b * 3 * cs + (size_t)x0 * W_ + y0;
            Ia[k] = make_float4(p[0],      p[cs],          p[2*cs],          0.f);
            Ib[k] = make_float4(p[1],      p[cs + 1],      p[2*cs + 1],      0.f);
            Ic[k] = make_float4(p[W_],     p[cs + W_],     p[2*cs + W_],     0.f);
            Id[k] = make_float4(p[W_ + 1], p[cs + W_ + 1], p[2*cs + W_ + 1], 0.f);
        }
    }

    // ---- phase 3: wait for DMA, then weighted accumulation from LDS ----
    wait_async_lds();
    __syncthreads();

    // neighbor sample 2n lives at pixel 2p: row ibase + (2j)/256, col (2j)%256
    const int j2 = tid << 1;
    const int rr = j2 >> 8;       // 0 or 1
    const int cc = j2 & 255;      // even, <= 254

    float acc0 = 0.f, acc1 = 0.f, acc2 = 0.f;
    #pragma unroll
    for (int k = 0; k < K2; ++k) {
        // samples m0=2n, m1=2n+1 -> tap (2k)%9 / (2k+1)%9, pixel +0/+1
        int t0 = 2 * k,     d0 = (t0 >= K2); t0 -= d0 * K2;
        int t1 = 2 * k + 1, d1 = (t1 >= K2); t1 -= d1 * K2;

        float a0 = sBuf[(9  + 2 * t0 + rr) * SEGF + cc + d0];
        float a1 = sBuf[(9  + 2 * t1 + rr) * SEGF + cc + d1];
        float b0 = sBuf[(27 + 2 * t0 + rr) * SEGF + cc + d0];
        float b1 = sBuf[(27 + 2 * t1 + rr) * SEGF + cc + d1];
        if (lower) { a0 = 1.f - a0; a1 = 1.f - a1; b0 = 1.f - b0; b1 = 1.f - b1; }

        const float kw  = sBuf[k * SEGF + tid];
        const float w00 = a0 * b0, w10 = a1 * b0, w01 = a0 * b1, w11 = a1 * b1;
        // reference's scrambled 2x2x3 corner/channel permutation
        acc0 += kw * (w00 * Ib[k].x + w10 * Ic[k].x + w01 * Id[k].y + w11 * Ia[k].z);
        acc1 += kw * (w00 * Ia[k].x + w10 * Ib[k].y + w01 * Ic[k].y + w11 * Id[k].z);
        acc2 += kw * (w00 * Id[k].x + w10 * Ia[k].y + w01 * Ib[k].z + w11 * Ic[k].z);
    }

    // softround(255 * acc) = y - sin(2*pi*y)/(2*pi)
    const float TWO_PI     = 6.2831853071795864769f;
    const float INV_TWO_PI = 0.15915494309189533577f;
    float r0 = 255.f * acc0, r1 = 255.f * acc1, r2 = 255.f * acc2;
    r0 -= __sinf(TWO_PI * r0) * INV_TWO_PI;
    r1 -= __sinf(TWO_PI * r1) * INV_TWO_PI;
    r2 -= __sinf(TWO_PI * r2) * INV_TWO_PI;

    float* op = out + (((size_t)b * h_ + i) * w_ + tid) * 3;
    __builtin_nontemporal_store(r0, op + 0);
    __builtin_nontemporal_store(r1, op + 1);
    __builtin_nontemporal_store(r2, op + 2);
}

// ---------------------------------------------------------------------------
extern "C" void kernel_launch(void* const* d_in, const int* in_sizes, int n_in,
                              void* d_out, int out_size, void* d_ws, size_t ws_size,
                              hipStream_t stream) {
    (void)in_sizes; (void)n_in; (void)out_size;
    const float* images  = (const float*)d_in[0];
    const float* kernels = (const float*)d_in[1];
    const float* offx    = (const float*)d_in[2];
    const float* offy    = (const float*)d_in[3];
    float* out = (float*)d_out;

    const size_t need = (size_t)B_ * H_ * W_ * sizeof(float4);   // 33.6 MB
    if (ws_size >= need && d_ws != nullptr) {
        float4* imcl = (float4*)d_ws;
        int n = B_ * H_ * W_;
        chw_to_hwc4<<<dim3((n + 255) / 256), dim3(256), 0, stream>>>(images, imcl);
        deform_gather<true><<<dim3(B_ * h_), dim3(256), 0, stream>>>(
            imcl, images, kernels, offx, offy, out);
    } else {
        deform_gather<false><<<dim3(B_ * h_), dim3(256), 0, stream>>>(
            nullptr, images, kernels, offx, offy, out);
    }
}